// LookaheadAdamBackprop_33844342292859
// MI455X (gfx1250) — compile-verified
//
#include <hip/hip_runtime.h>

// Lookahead-AdamW fused update, MI455X (gfx1250, wave32).
// Pure streaming kernel: 36 B/element of HBM traffic (1.15 GB for N=32M),
// ~50us floor @ 23.3 TB/s. Strategy:
//   - b128 non-temporal loads/stores (streamed data can't fit 192MB L2)
//   - global_prefetch_b8 one grid-stride (16MB/stream) ahead into GL2
//   - step-dependent scalars precomputed by a 1-thread setup kernel into d_ws
//     so hot waves start streaming immediately (no per-wave powf expansion)

typedef float v4f __attribute__((ext_vector_type(4)));

#define K_STEP_SIZE   0.001f
#define K_BETA1       0.9f
#define K_BETA2       0.999f
#define K_EPS         1e-8f
#define K_WD          0.01f
#define K_SYNC_PERIOD 5
#define K_SLOW_STEP   0.5f

// ws[0] = 1/(1-beta1^step), ws[1] = 1/(1-beta2^step), ws[2] = sync flag (bits)
__global__ void la_adamw_setup(const int* __restrict__ step_ptr,
                               float* __restrict__ ws) {
    if (threadIdx.x == 0 && blockIdx.x == 0) {
        const int step = *step_ptr;
        ws[0] = 1.0f / (1.0f - __powf(K_BETA1, (float)step));
        ws[1] = 1.0f / (1.0f - __powf(K_BETA2, (float)step));
        ws[2] = ((step % K_SYNC_PERIOD) == 0) ? 1.0f : 0.0f;
    }
}

__global__ __launch_bounds__(256) void la_adamw_v4(
    const v4f* __restrict__ param,
    const v4f* __restrict__ grad,
    const v4f* __restrict__ m_in,
    const v4f* __restrict__ v_in,
    const v4f* __restrict__ slow,
    const float* __restrict__ ws,
    v4f* __restrict__ fast_out,
    v4f* __restrict__ m_out,
    v4f* __restrict__ v_out,
    v4f* __restrict__ slow_out,
    int n4)
{
    // Uniform scalars precomputed by setup kernel: three s_load dwords.
    const float inv_bc1 = ws[0];
    const float inv_bc2 = ws[1];
    const bool  sync    = ws[2] != 0.0f;

    const int stride = (int)(gridDim.x * blockDim.x);
    for (int i = (int)(blockIdx.x * blockDim.x + threadIdx.x); i < n4; i += stride) {
        // Prefetch next grid-stride tile into GL2 while this one is in flight.
        const int nx = i + stride;
        if (nx < n4) {
            __builtin_prefetch(param + nx, 0, 0);
            __builtin_prefetch(grad  + nx, 0, 0);
            __builtin_prefetch(m_in  + nx, 0, 0);
            __builtin_prefetch(v_in  + nx, 0, 0);
            __builtin_prefetch(slow  + nx, 0, 0);
        }

        // Streaming data: non-temporal b128 loads (no reuse, don't pollute L2).
        const v4f p  = __builtin_nontemporal_load(param + i);
        const v4f g  = __builtin_nontemporal_load(grad  + i);
        const v4f mm = __builtin_nontemporal_load(m_in  + i);
        const v4f vv = __builtin_nontemporal_load(v_in  + i);
        const v4f s  = __builtin_nontemporal_load(slow  + i);

        // AdamW: decay folded into grad (per reference).
        const v4f gw = g + K_WD * p;
        const v4f mn = K_BETA1 * mm + (1.0f - K_BETA1) * gw;
        const v4f vn = K_BETA2 * vv + (1.0f - K_BETA2) * (gw * gw);

        v4f fast;
        #pragma unroll
        for (int k = 0; k < 4; ++k) {
            const float mh = mn[k] * inv_bc1;
            const float vh = vn[k] * inv_bc2;
            fast[k] = p[k] - K_STEP_SIZE * mh / (__builtin_sqrtf(vh) + K_EPS);
        }

        v4f snew = s;
        if (sync) {  // uniform scalar branch
            #pragma unroll
            for (int k = 0; k < 4; ++k)
                snew[k] = s[k] + K_SLOW_STEP * (fast[k] - s[k]);
            fast = snew;
        }

        __builtin_nontemporal_store(fast, fast_out + i);
        __builtin_nontemporal_store(mn,   m_out   + i);
        __builtin_nontemporal_store(vn,   v_out   + i);
        __builtin_nontemporal_store(snew, slow_out + i);
    }
}

// Scalar tail for N % 4 != 0 (not hit for N = 32M, kept for safety).
__global__ __launch_bounds__(64) void la_adamw_tail(
    const float* __restrict__ param,
    const float* __restrict__ grad,
    const float* __restrict__ m_in,
    const float* __restrict__ v_in,
    const float* __restrict__ slow,
    const float* __restrict__ ws,
    float* __restrict__ fast_out,
    float* __restrict__ m_out,
    float* __restrict__ v_out,
    float* __restrict__ slow_out,
    int base, int n)
{
    const float inv_bc1 = ws[0];
    const float inv_bc2 = ws[1];
    const bool  sync    = ws[2] != 0.0f;

    const int i = base + (int)(blockIdx.x * blockDim.x + threadIdx.x);
    if (i >= n) return;

    const float p  = param[i];
    const float gw = grad[i] + K_WD * p;
    const float mn = K_BETA1 * m_in[i] + (1.0f - K_BETA1) * gw;
    const float vn = K_BETA2 * v_in[i] + (1.0f - K_BETA2) * (gw * gw);
    float fast = p - K_STEP_SIZE * (mn * inv_bc1) /
                     (__builtin_sqrtf(vn * inv_bc2) + K_EPS);
    const float s = slow[i];
    float snew = s;
    if (sync) { snew = s + K_SLOW_STEP * (fast - s); fast = snew; }
    fast_out[i] = fast;
    m_out[i]    = mn;
    v_out[i]    = vn;
    slow_out[i] = snew;
}

extern "C" void kernel_launch(void* const* d_in, const int* in_sizes, int n_in,
                              void* d_out, int out_size, void* d_ws, size_t ws_size,
                              hipStream_t stream) {
    (void)n_in; (void)ws_size; (void)out_size;

    const float* param = (const float*)d_in[0];
    const float* grad  = (const float*)d_in[1];
    const float* m_in  = (const float*)d_in[2];
    const float* v_in  = (const float*)d_in[3];
    const float* slow  = (const float*)d_in[4];
    const int*   step  = (const int*)d_in[5];
    float*       ws    = (float*)d_ws;

    const int n  = in_sizes[0];
    const int n4 = n / 4;
    const int rem_base = n4 * 4;

    float* out      = (float*)d_out;
    float* fast_out = out;
    float* m_out    = out + (size_t)n;
    float* v_out    = out + 2 * (size_t)n;
    float* slow_out = out + 3 * (size_t)n;

    // Stage 1: fold step -> {inv_bc1, inv_bc2, sync} once (same-stream order
    // guarantees visibility to the streaming kernels, graph-capture safe).
    la_adamw_setup<<<1, 32, 0, stream>>>(step, ws);

    if (n4 > 0) {
        // Persistent-style grid: ~8 float4 iterations/thread so the
        // one-stride-ahead prefetch has real work to overlap with.
        int blocks = 4096;
        const int max_blocks = (n4 + 255) / 256;
        if (blocks > max_blocks) blocks = max_blocks;
        la_adamw_v4<<<blocks, 256, 0, stream>>>(
            (const v4f*)param, (const v4f*)grad, (const v4f*)m_in,
            (const v4f*)v_in, (const v4f*)slow, ws,
            (v4f*)fast_out, (v4f*)m_out, (v4f*)v_out, (v4f*)slow_out, n4);
    }
    if (rem_base < n) {
        const int tail = n - rem_base;
        la_adamw_tail<<<(tail + 63) / 64, 64, 0, stream>>>(
            param, grad, m_in, v_in, slow, ws,
            fast_out, m_out, v_out, slow_out, rem_base, n);
    }
}